// TexRender_18726057411148
// MI455X (gfx1250) — compile-verified
//
#include <hip/hip_runtime.h>
#include <math.h>

typedef float v2f __attribute__((ext_vector_type(2)));
typedef float v8f __attribute__((ext_vector_type(8)));
typedef unsigned int u32x4 __attribute__((ext_vector_type(4)));
typedef int i32x4 __attribute__((ext_vector_type(4)));
typedef int i32x8 __attribute__((ext_vector_type(8)));

#define FPAD_MAX 1008  // F=1000 padded to 16; 16*1008*4B = 64512B LDS blob

// D = A(16x4) * B(4x16) + 0, f32 WMMA (CDNA5 wave32).
static __device__ __forceinline__ v8f wmma4(v2f a, v2f b) {
  v8f c = {};
  return __builtin_amdgcn_wmma_f32_16x16x4_f32(false, a, false, b, (short)0, c,
                                               false, false);
}

// TDM: 1-D DMA of `nwords` dwords from global -> LDS offset `lds_off`.
// D# per CDNA5 ISA ch.8: group0 = {count|flags, lds_addr, global_addr, type=2},
// group1 = {data_size=4B, tensor_dim0, tile_dim0, tensor_dim0_stride}, 1-D tile.
static __device__ __forceinline__ void tdm_load_1d(unsigned lds_off,
                                                   const float* gsrc,
                                                   unsigned nwords) {
  unsigned long long ga = (unsigned long long)(uintptr_t)gsrc;
  u32x4 g0;
  g0[0] = 1u;  // count=1, is_restore=0, gather_mode=0
  g0[1] = lds_off;
  g0[2] = (unsigned)(ga & 0xFFFFFFFFu);
  g0[3] = (unsigned)((ga >> 32) & 0x01FFFFFFu) | (2u << 30);  // type=2 (image)
  i32x8 g1;
  g1[0] = (int)(2u << 16);                  // data_size=2 -> 4 bytes
  g1[1] = (int)((nwords & 0xFFFFu) << 16);  // tensor_dim0[15:0]
  g1[2] = (int)(((nwords >> 16) & 0xFFFFu) | (1u << 16));  // dim0 hi | dim1=1
  g1[3] = (int)((nwords & 0xFFFFu) << 16);  // tile_dim0 = nwords
  g1[4] = 0;                                // tile_dim1/2 = 0 (1-D)
  g1[5] = (int)nwords;                      // tensor_dim0_stride lo
  g1[6] = 0;
  g1[7] = 0;
  i32x4 z4 = {};
#if __clang_major__ >= 23
  i32x8 z8 = {};
  __builtin_amdgcn_tensor_load_to_lds(g0, g1, z4, z4, z8, 0);
#else
  __builtin_amdgcn_tensor_load_to_lds(g0, g1, z4, z4, 0);
#endif
}

// ---------------- Kernel 1: point transform + projection ----------------
__global__ void geom_points(const float* __restrict__ pts_in,
                            const float* __restrict__ rot,
                            const float* __restrict__ campos,
                            const float* __restrict__ proj,
                            float* __restrict__ pts_cam,
                            float* __restrict__ xy, int B, int P) {
  int idx = blockIdx.x * blockDim.x + threadIdx.x;
  if (idx >= B * P) return;
  int b = idx / P;
  const float* Rb = rot + b * 9;
  const float* cb = campos + b * 3;
  float dx = pts_in[idx * 3 + 0] - cb[0];
  float dy = pts_in[idx * 3 + 1] - cb[1];
  float dz = pts_in[idx * 3 + 2] - cb[2];
  float p0 = Rb[0] * dx + Rb[1] * dy + Rb[2] * dz;
  float p1 = Rb[3] * dx + Rb[4] * dy + Rb[5] * dz;
  float p2 = Rb[6] * dx + Rb[7] * dy + Rb[8] * dz;
  pts_cam[idx * 3 + 0] = p0;
  pts_cam[idx * 3 + 1] = p1;
  pts_cam[idx * 3 + 2] = p2;
  float z3 = p2 * proj[2];
  xy[idx * 2 + 0] = p0 * proj[0] / z3;
  xy[idx * 2 + 1] = p1 * proj[1] / z3;
}

// ---------------- Kernel 2: per-face setup into contiguous blob ----------------
// blob[b] rows (each Fpad floats): 0..8 = coef[i*3+k], 9..11 = z-reduction
// coefs, 12..14 = denom-reduction coefs, 15 = front flag.
__global__ void geom_faces(const int* __restrict__ faces,
                           const int* __restrict__ ft,
                           const float* __restrict__ pts_cam,
                           const float* __restrict__ xy,
                           const float* __restrict__ uv,
                           float* __restrict__ blob, float* __restrict__ fu,
                           float* __restrict__ fv,
                           float* __restrict__ out_normal, int B, int P, int F,
                           int Fpad) {
  int idx = blockIdx.x * blockDim.x + threadIdx.x;
  if (idx >= B * Fpad) return;
  int b = idx / Fpad, f = idx % Fpad;
  float* Lb = blob + (size_t)b * 16 * Fpad;
  float* fub = fu + b * 3 * Fpad;
  float* fvb = fv + b * 3 * Fpad;
  if (f >= F) {  // padding face: inert (front=0 -> never valid, soft=0)
#pragma unroll
    for (int i = 0; i < 16; ++i) Lb[i * Fpad + f] = 0.f;
#pragma unroll
    for (int k = 0; k < 3; ++k) {
      fub[k * Fpad + f] = 0.f;
      fvb[k * Fpad + f] = 0.f;
    }
    return;
  }
  int i0 = faces[f * 3 + 0], i1 = faces[f * 3 + 1], i2 = faces[f * 3 + 2];
  const float* pb = pts_cam + b * P * 3;
  const float* xyb = xy + b * P * 2;
  float p0z = pb[i0 * 3 + 2];
  float p1z = pb[i1 * 3 + 2];
  float p2z = pb[i2 * 3 + 2];
  float x0 = xyb[i0 * 2], y0 = xyb[i0 * 2 + 1];
  float x1 = xyb[i1 * 2], y1 = xyb[i1 * 2 + 1];
  float x2 = xyb[i2 * 2], y2 = xyb[i2 * 2 + 1];
  float c[3][3];
  c[0][0] = y1 - y2; c[0][1] = x2 - x1; c[0][2] = x1 * y2 - x2 * y1;
  c[1][0] = y2 - y0; c[1][1] = x0 - x2; c[1][2] = x2 * y0 - x0 * y2;
  c[2][0] = y0 - y1; c[2][1] = x1 - x0; c[2][2] = x0 * y1 - x1 * y0;
#pragma unroll
  for (int i = 0; i < 3; ++i)
#pragma unroll
    for (int k = 0; k < 3; ++k) Lb[(i * 3 + k) * Fpad + f] = c[i][k];
#pragma unroll
  for (int k = 0; k < 3; ++k) {
    Lb[(9 + k) * Fpad + f] = c[0][k] * p0z + c[1][k] * p1z + c[2][k] * p2z;
    Lb[(12 + k) * Fpad + f] = c[0][k] + c[1][k] + c[2][k];
  }
  float p0x = pb[i0 * 3], p0y = pb[i0 * 3 + 1];
  float e1x = pb[i1 * 3] - p0x, e1y = pb[i1 * 3 + 1] - p0y, e1z = p1z - p0z;
  float e2x = pb[i2 * 3] - p0x, e2y = pb[i2 * 3 + 1] - p0y, e2z = p2z - p0z;
  float nx = e1y * e2z - e1z * e2y;
  float ny = e1z * e2x - e1x * e2z;
  float nz = e1x * e2y - e1y * e2x;
  Lb[15 * Fpad + f] = (nz > 0.f) ? 1.f : 0.f;
  float nlen = sqrtf(nx * nx + ny * ny + nz * nz + 1e-8f);
  float ninv = 1.f / (nlen + 1e-15f);
  out_normal[(b * F + f) * 3 + 0] = nx * ninv;
  out_normal[(b * F + f) * 3 + 1] = ny * ninv;
  out_normal[(b * F + f) * 3 + 2] = nz * ninv;
  int t0 = ft[f * 3 + 0], t1 = ft[f * 3 + 1], t2 = ft[f * 3 + 2];
  const float* uvb = uv + b * P * 2;
  fub[0 * Fpad + f] = uvb[t0 * 2]; fvb[0 * Fpad + f] = uvb[t0 * 2 + 1];
  fub[1 * Fpad + f] = uvb[t1 * 2]; fvb[1 * Fpad + f] = uvb[t1 * 2 + 1];
  fub[2 * Fpad + f] = uvb[t2 * 2]; fvb[2 * Fpad + f] = uvb[t2 * 2 + 1];
}

// ---------------- Kernel 3: TDM->LDS stage + WMMA rasterizer ----------------
__global__ void __launch_bounds__(256) raster_kernel(
    const float* __restrict__ blob, const float* __restrict__ fu,
    const float* __restrict__ fv, const float* __restrict__ tex,
    float* __restrict__ out, int B, int F, int Fpad, int H, int W, int TH,
    int TW) {
  __shared__ float L[16 * FPAD_MAX];  // 64512 B: whole face table for batch b

  const int lane = threadIdx.x & 31;
  const int wave = threadIdx.x >> 5;
  const int stripsPerB = (H * W) >> 4;
  const int strip = blockIdx.x * 8 + wave;
  const bool live = strip < B * stripsPerB;
  const int b = live ? (strip / stripsPerB) : 0;

  // One TDM DMA per block stages the entire per-batch face blob into LDS.
  if (wave == 0) {
    unsigned lds_off = (unsigned)(uintptr_t)(void*)L;
    tdm_load_1d(lds_off, blob + (size_t)b * 16 * Fpad, 16u * (unsigned)Fpad);
    __builtin_amdgcn_s_wait_tensorcnt(0);
  }
  __syncthreads();
  if (!live) return;  // wave-uniform

  const int pbase = (strip % stripsPerB) << 4;
  const int m = lane & 15;
  const bool hi = lane >= 16;
  const int pxi = pbase + m;
  const int axx = pxi % W, ayy = pxi / W;
  const float px = (axx + 0.5f) / W * 2.f - 1.f;
  const float py = 1.f - (ayy + 0.5f) / H * 2.f;
  v2f A;
  A.x = hi ? 1.f : px;  // K2 : K0
  A.y = hi ? 0.f : py;  // K3 : K1

  float bestz[8], msoft[8];
  int bestf[8];
  int covmask = 0;
#pragma unroll
  for (int r = 0; r < 8; ++r) {
    bestz[r] = INFINITY;
    bestf[r] = 0;
    msoft[r] = 0.f;
  }

  const int kA = hi ? 2 : 0;
  const float ymul = hi ? 0.f : 1.f;
  for (int c0 = 0; c0 < Fpad; c0 += 16) {
    const int f = c0 + m;          // this lane's face column
    const int ex = f + kA * Fpad;  // base element for K=kA rows
    const int ey = f + Fpad;       // base element for K=1 rows
    // All ten ds_loads are unconditional (shared bases + constant offsets);
    // hi-lane .y components are zeroed by multiplying with ymul, never a branch.
    v2f B0, B1, B2, Bz, Bd;
    B0.x = L[ex + 0 * Fpad];
    B1.x = L[ex + 3 * Fpad];
    B2.x = L[ex + 6 * Fpad];
    Bz.x = L[ex + 9 * Fpad];
    Bd.x = L[ex + 12 * Fpad];
    float y0 = L[ey + 0 * Fpad];
    float y1 = L[ey + 3 * Fpad];
    float y2 = L[ey + 6 * Fpad];
    float yz = L[ey + 9 * Fpad];
    float yd = L[ey + 12 * Fpad];
    B0.y = y0 * ymul;
    B1.y = y1 * ymul;
    B2.y = y2 * ymul;
    Bz.y = yz * ymul;
    Bd.y = yd * ymul;
    const float frf = L[f + 15 * Fpad];  // exactly 1.0 or 0.0
    const bool front = frf > 0.f;

    v8f W0 = wmma4(A, B0);
    v8f W1 = wmma4(A, B1);
    v8f W2 = wmma4(A, B2);
    v8f Wz = wmma4(A, Bz);
    v8f Wd = wmma4(A, Bd);

#pragma unroll
    for (int r = 0; r < 8; ++r) {
      float w0 = W0[r], w1 = W1[r], w2 = W2[r];
      float den = Wd[r];
      float dadj = den + ((den >= 0.f) ? 1e-8f : -1e-8f);
      float inv = __builtin_amdgcn_rcpf(dadj);  // |dadj| >= ~1e-8, safe
      float b0 = w0 * inv, b1 = w1 * inv, b2 = w2 * inv;
      bool inside = (b0 >= 0.f) & (b1 >= 0.f) & (b2 >= 0.f);
      bool valid = inside & front;
      float zp = Wz[r] * inv;
      bool upd = valid & (zp < bestz[r]);  // branchless -> v_cndmask
      bestz[r] = upd ? zp : bestz[r];
      bestf[r] = upd ? f : bestf[r];
      covmask |= ((int)valid) << r;
      float dmin = fminf(b0, fminf(b1, b2));  // v_min3
      // sigmoid via fast rcp; frf in {0,1} makes the front-select branchless
      float soft = __builtin_amdgcn_rcpf(1.f + __expf(dmin * -50.f)) * frf;
      msoft[r] = fmaxf(msoft[r], soft);
    }
  }

  // reduce across the 16 face-lanes of each half (xor 1,2,4,8 stays in-half)
#pragma unroll
  for (int off = 1; off < 16; off <<= 1) {
#pragma unroll
    for (int r = 0; r < 8; ++r) {
      float oz = __shfl_xor(bestz[r], off);
      int ofc = __shfl_xor(bestf[r], off);
      bool upd = oz < bestz[r];
      bestz[r] = upd ? oz : bestz[r];
      bestf[r] = upd ? ofc : bestf[r];
      msoft[r] = fmaxf(msoft[r], __shfl_xor(msoft[r], off));
    }
    covmask |= __shfl_xor(covmask, off);
  }

  // finalize: lanes 0-7 write pixels 0-7, lanes 16-23 write pixels 8-15
  bool writer = (!hi && lane < 8) || (hi && lane < 24);
  if (writer) {
    int r = lane & 7;
    int pm = pbase + (hi ? 8 : 0) + r;
    int xx = pm % W, yy = pm / W;
    float fpx = (xx + 0.5f) / W * 2.f - 1.f;
    float fpy = 1.f - (yy + 0.5f) / H * 2.f;
    float u = 0.f, v = 0.f, maskv = 0.f;
    if ((covmask >> r) & 1) {
      int f = bestf[r];
      float wv[3];
#pragma unroll
      for (int i = 0; i < 3; ++i)
        wv[i] = L[(i * 3 + 0) * Fpad + f] * fpx +
                L[(i * 3 + 1) * Fpad + f] * fpy + L[(i * 3 + 2) * Fpad + f];
      float den = wv[0] + wv[1] + wv[2];
      float dadj = den + ((den >= 0.f) ? 1e-8f : -1e-8f);
      float inv = __builtin_amdgcn_rcpf(dadj);
#pragma unroll
      for (int i = 0; i < 3; ++i) {
        float bi = wv[i] * inv;
        u += bi * fu[(b * 3 + i) * Fpad + f];
        v += bi * fv[(b * 3 + i) * Fpad + f];
        maskv += bi;
      }
    }
    // fragment shader: wrap, flip y, nearest sample, zeros padding
    float gu = u - floorf(u);
    float gv = v - floorf(v);
    float gx = gu * 2.f - 1.f;
    float gy = -(gv * 2.f - 1.f);
    float ix = ((gx + 1.f) * TW - 1.f) * 0.5f;
    float iy = ((gy + 1.f) * TH - 1.f) * 0.5f;
    int ixn = (int)rintf(ix);  // round-half-even == jnp.round
    int iyn = (int)rintf(iy);
    bool vld = (ixn >= 0) && (ixn < TW) && (iyn >= 0) && (iyn < TH);
    int ixc = min(max(ixn, 0), TW - 1);
    int iyc = min(max(iyn, 0), TH - 1);
    int improbOff = B * H * W * 3;
#pragma unroll
    for (int ch = 0; ch < 3; ++ch) {
      float t = vld ? tex[((b * 3 + ch) * TH + iyc) * TW + ixc] : 0.f;
      out[((b * H + yy) * W + xx) * 3 + ch] =
          fminf(fmaxf(t * maskv, 0.f), 1.f);
    }
    out[improbOff + (b * H + yy) * W + xx] = msoft[r];
  }
}

extern "C" void kernel_launch(void* const* d_in, const int* in_sizes, int n_in,
                              void* d_out, int out_size, void* d_ws,
                              size_t ws_size, hipStream_t stream) {
  const float* points = (const float*)d_in[0];
  const int* faces = (const int*)d_in[1];
  const float* rot = (const float*)d_in[2];
  const float* campos = (const float*)d_in[3];
  const float* proj = (const float*)d_in[4];
  const float* uv = (const float*)d_in[5];
  const float* tex = (const float*)d_in[6];
  const int* ft = (const int*)d_in[7];

  int B = in_sizes[2] / 9;
  int F = in_sizes[1] / 3;
  int P = in_sizes[0] / (3 * B);
  int THW = in_sizes[6] / (3 * B);
  int TH = (int)(sqrt((double)THW) + 0.5);
  int TW = TH;
  int HW = (out_size - 3 * B * F) / (4 * B);
  int H = (int)(sqrt((double)HW) + 0.5);
  int W = H;
  int Fpad = (F + 15) & ~15;
  if (Fpad > FPAD_MAX) Fpad = FPAD_MAX;  // LDS blob sizing assumption

  float* ws = (float*)d_ws;
  float* pts_cam = ws;  ws += (size_t)B * P * 3;
  float* xy = ws;       ws += (size_t)B * P * 2;
  float* blob = ws;     ws += (size_t)B * 16 * Fpad;
  float* fu = ws;       ws += (size_t)B * 3 * Fpad;
  float* fv = ws;

  float* out = (float*)d_out;
  float* out_normal = out + (size_t)B * H * W * 4;

  int nPts = B * P;
  geom_points<<<(nPts + 255) / 256, 256, 0, stream>>>(points, rot, campos,
                                                      proj, pts_cam, xy, B, P);
  int nF = B * Fpad;
  geom_faces<<<(nF + 255) / 256, 256, 0, stream>>>(
      faces, ft, pts_cam, xy, uv, blob, fu, fv, out_normal, B, P, F, Fpad);
  int strips = B * ((H * W) >> 4);
  int blocks = (strips + 7) / 8;  // 8 waves of 32 per block
  raster_kernel<<<blocks, 256, 0, stream>>>(blob, fu, fv, tex, out, B, F, Fpad,
                                            H, W, TH, TW);
}